// CFGATLayer_1872605741300
// MI455X (gfx1250) — compile-verified
//
#include <hip/hip_runtime.h>

#define B_ 4
#define N_ 4096
#define D_ 128
#define ALPHA_ 0.2f
#define MASK_VAL_ -9e15f
#define LN_EPS_ 1e-5f

typedef float v2f __attribute__((ext_vector_type(2)));
typedef float v8f __attribute__((ext_vector_type(8)));

__device__ __forceinline__ float leaky(float x) { return x >= 0.f ? x : ALPHA_ * x; }

__device__ __forceinline__ float wave_sum(float v) {
#pragma unroll
    for (int off = 16; off >= 1; off >>= 1) v += __shfl_xor(v, off);
    return v;
}

// ---------------------------------------------------------------------------
// Kernel 1: fused node-affine + residual, h = x@W, LayerNorm, f1 = h@a1,
// f2 = h@a2.  One wave32 per row; lane owns D elements [4*lane .. 4*lane+3].
// W (64 KB) streams through WGP$/L2 with fully coalesced b128 loads.
// ---------------------------------------------------------------------------
__global__ void __launch_bounds__(256) gat_prelude_kernel(
    const float* __restrict__ x, const float* __restrict__ W,
    const float* __restrict__ a1, const float* __restrict__ a2,
    const float* __restrict__ nw, const float* __restrict__ nb,
    const float* __restrict__ gamma, const float* __restrict__ beta,
    float* __restrict__ hln, float* __restrict__ f1, float* __restrict__ f2)
{
    const int lane   = threadIdx.x & 31;
    const int wave   = blockIdx.x * (blockDim.x >> 5) + (threadIdx.x >> 5);
    const int nWaves = gridDim.x * (blockDim.x >> 5);

    for (int row = wave; row < B_ * N_; row += nWaves) {
        const int n = row & (N_ - 1);
        // affine + residual: x*(nw+1) + nb
        float4 xv = *(const float4*)(x + (size_t)row * D_ + 4 * lane);
        const float s  = nw[n] + 1.0f;
        const float bb = nb[n];
        xv.x = xv.x * s + bb; xv.y = xv.y * s + bb;
        xv.z = xv.z * s + bb; xv.w = xv.w * s + bb;

        // h[d] = sum_k xr[k] * W[k][d]; broadcast xr[k] with shfl
        float4 h = {0.f, 0.f, 0.f, 0.f};
#pragma unroll
        for (int comp = 0; comp < 4; ++comp) {
            const float xc = (comp == 0) ? xv.x : (comp == 1) ? xv.y
                           : (comp == 2) ? xv.z : xv.w;
#pragma unroll 8
            for (int l = 0; l < 32; ++l) {
                const float xk = __shfl(xc, l);
                const int   k  = 4 * l + comp;
                const float4 w4 = *(const float4*)(W + (size_t)k * D_ + 4 * lane);
                h.x += xk * w4.x; h.y += xk * w4.y;
                h.z += xk * w4.z; h.w += xk * w4.w;
            }
        }

        // LayerNorm over D=128
        const float mu = wave_sum(h.x + h.y + h.z + h.w) * (1.0f / 128.0f);
        const float dx = h.x - mu, dy = h.y - mu, dz = h.z - mu, dw = h.w - mu;
        const float var = wave_sum(dx * dx + dy * dy + dz * dz + dw * dw) * (1.0f / 128.0f);
        const float rstd = rsqrtf(var + LN_EPS_);
        const float4 g  = *(const float4*)(gamma + 4 * lane);
        const float4 be = *(const float4*)(beta  + 4 * lane);
        float4 hn;
        hn.x = dx * rstd * g.x + be.x;
        hn.y = dy * rstd * g.y + be.y;
        hn.z = dz * rstd * g.z + be.z;
        hn.w = dw * rstd * g.w + be.w;

        // attention score projections
        const float4 A1 = *(const float4*)(a1 + 4 * lane);
        const float4 A2 = *(const float4*)(a2 + 4 * lane);
        const float p1 = wave_sum(hn.x * A1.x + hn.y * A1.y + hn.z * A1.z + hn.w * A1.w);
        const float p2 = wave_sum(hn.x * A2.x + hn.y * A2.y + hn.z * A2.z + hn.w * A2.w);

        *(float4*)(hln + (size_t)row * D_ + 4 * lane) = hn;
        if (lane == 0) { f1[row] = p1; f2[row] = p2; }
    }
}

// ---------------------------------------------------------------------------
// Kernel 2: h' = att @ h with att built on the fly, fp32 WMMA 16x16x4.
// Each wave computes a 16-row x 64-col output tile (4 C fragments).
// h is staged per 64-row K-chunk in LDS (stride 132 to avoid bank conflicts),
// shared by all 8 waves of the block (same batch).
// A fragment layout (16x4 f32): lanes 0-15 hold K0/K1, lanes 16-31 hold K2/K3.
// ---------------------------------------------------------------------------
__global__ void __launch_bounds__(256) gat_attn_wmma_kernel(
    const int*   __restrict__ adj, const float* __restrict__ hln,
    const float* __restrict__ f1,  const float* __restrict__ f2,
    float* __restrict__ out)
{
    __shared__ float hs[64 * 132];
    __shared__ float f2s[64];

    const int tid    = threadIdx.x;
    const int lane   = tid & 31;
    const int waveId = tid >> 5;
    const int b      = blockIdx.x >> 6;                  // 64 blocks per batch
    const int task   = ((blockIdx.x & 63) << 3) + waveId; // 0..511
    const int i0     = (task >> 1) << 4;                 // row strip base
    const int n0     = (task & 1) << 6;                  // column half base
    const int m      = lane & 15;
    const int p      = lane >> 4;

    const float f1v = f1[b * N_ + i0 + m];
    const float* hb  = hln + (size_t)b * N_ * D_;
    const float* f2b = f2 + b * N_;
    const int* adjRow = adj + (size_t)(i0 + m) * N_;

    v8f acc0 = {}, acc1 = {}, acc2 = {}, acc3 = {};

    for (int kb = 0; kb < N_; kb += 64) {
        __syncthreads();
        // cooperative stage of h[kb..kb+63][0..127] into LDS (float4, padded rows)
#pragma unroll
        for (int r = 0; r < 8; ++r) {
            const int idx  = tid + r * 256;      // 0..2047 float4 slots
            const int krow = idx >> 5;
            const int c4   = (idx & 31) << 2;
            const float4 v = *(const float4*)(hb + (size_t)(kb + krow) * D_ + c4);
            *(float4*)(&hs[krow * 132 + c4]) = v;
        }
        if (tid < 64) f2s[tid] = f2b[kb + tid];
        __syncthreads();

#pragma unroll 4
        for (int kk = 0; kk < 64; kk += 4) {
            const int kA = kk + p * 2;
            // A fragment: att[i0+m, kb+kA .. +1] built on the fly
            const int2 ad = *(const int2*)(adjRow + kb + kA);
            const float e0 = leaky(f1v + f2s[kA]);
            const float e1 = leaky(f1v + f2s[kA + 1]);
            const float a0 = leaky(ad.x > 0 ? e0 : MASK_VAL_);
            const float a1v = leaky(ad.y > 0 ? e1 : MASK_VAL_);
            v2f Av; Av[0] = a0; Av[1] = a1v;

            const float* h0 = &hs[kA * 132 + n0 + m];
            const float* h1 = &hs[(kA + 1) * 132 + n0 + m];
            v2f B0; B0[0] = h0[0];  B0[1] = h1[0];
            v2f B1; B1[0] = h0[16]; B1[1] = h1[16];
            v2f B2; B2[0] = h0[32]; B2[1] = h1[32];
            v2f B3; B3[0] = h0[48]; B3[1] = h1[48];

            acc0 = __builtin_amdgcn_wmma_f32_16x16x4_f32(false, Av, false, B0, (short)0, acc0, false, false);
            acc1 = __builtin_amdgcn_wmma_f32_16x16x4_f32(false, Av, false, B1, (short)0, acc1, false, false);
            acc2 = __builtin_amdgcn_wmma_f32_16x16x4_f32(false, Av, false, B2, (short)0, acc2, false, false);
            acc3 = __builtin_amdgcn_wmma_f32_16x16x4_f32(false, Av, false, B3, (short)0, acc3, false, false);
        }
    }

    // epilogue: out = elu(acc) + h_ln   (C layout: VGPR v -> M = p*8+v, N = lane%16)
#pragma unroll
    for (int t = 0; t < 4; ++t) {
        const v8f a = (t == 0) ? acc0 : (t == 1) ? acc1 : (t == 2) ? acc2 : acc3;
        const int col = n0 + t * 16 + m;
#pragma unroll
        for (int v = 0; v < 8; ++v) {
            const int i = i0 + p * 8 + v;
            const float c = a[v];
            const float o = (c > 0.f) ? c : (__expf(c) - 1.0f);
            out[(size_t)(b * N_ + i) * D_ + col] = o + hb[(size_t)i * D_ + col];
        }
    }
}

extern "C" void kernel_launch(void* const* d_in, const int* in_sizes, int n_in,
                              void* d_out, int out_size, void* d_ws, size_t ws_size,
                              hipStream_t stream) {
    const float* x     = (const float*)d_in[0];
    const int*   adj   = (const int*)d_in[1];
    const float* W     = (const float*)d_in[2];
    const float* a1    = (const float*)d_in[3];
    const float* a2    = (const float*)d_in[4];
    const float* nw    = (const float*)d_in[5];
    const float* nb    = (const float*)d_in[6];
    const float* gamma = (const float*)d_in[7];
    const float* beta  = (const float*)d_in[8];
    float* out = (float*)d_out;

    float* hln = (float*)d_ws;                 // B*N*D floats = 8 MB
    float* f1  = hln + (size_t)B_ * N_ * D_;   // B*N floats
    float* f2  = f1 + (size_t)B_ * N_;         // B*N floats

    gat_prelude_kernel<<<256, 256, 0, stream>>>(x, W, a1, a2, nw, nb, gamma, beta,
                                                hln, f1, f2);
    gat_attn_wmma_kernel<<<256, 256, 0, stream>>>(adj, hln, f1, f2, out);
}